// GCN_27702539059304
// MI455X (gfx1250) — compile-verified
//
#include <hip/hip_runtime.h>

typedef float v2f __attribute__((ext_vector_type(2)));
typedef float v8f __attribute__((ext_vector_type(8)));

#define NN     8192
#define BATCH  8
#define CIN    64
#define COUT   64
#define CHID   32

// -------------------------------------------------------------------------
// Inner 64-wide K-chunk for kernel 1.
// Wave-private 32x32 output tile: rows {ml, ml+16} x col-tiles {c0, c1}.
// STR is compile-time so all B addresses fold into instruction offsets:
//   b = cp[ s*4*STR + {0,1}*STR ]  -> global_load_b32 vaddr, offset:imm
// cp0/cp1 already include the per-lane offset (kl*STR + ml) and advance by
// one chunk (64*STR floats) per call.
// -------------------------------------------------------------------------
template<int STR>
__device__ __forceinline__ void gemm_chunk(
    const float* adjS,                 // LDS: 32 rows x 66 floats
    const float*& cp0, const float*& cp1,
    int ml, int kl,
    v8f& a00, v8f& a01, v8f& a10, v8f& a11)
{
#pragma unroll
    for (int s = 0; s < 16; ++s) {
        const int k4 = s * 4;
        v2f aA = *(const v2f*)(&adjS[ml * 66 + k4 + kl]);          // rows 0..15
        v2f aB = *(const v2f*)(&adjS[(ml + 16) * 66 + k4 + kl]);   // rows 16..31
        v2f b0, b1;
        b0.x = cp0[k4 * STR];
        b0.y = cp0[(k4 + 1) * STR];
        b1.x = cp1[k4 * STR];
        b1.y = cp1[(k4 + 1) * STR];
        a00 = __builtin_amdgcn_wmma_f32_16x16x4_f32(false, aA, false, b0,
                                                    (short)0, a00, false, false);
        a01 = __builtin_amdgcn_wmma_f32_16x16x4_f32(false, aA, false, b1,
                                                    (short)0, a01, false, false);
        a10 = __builtin_amdgcn_wmma_f32_16x16x4_f32(false, aB, false, b0,
                                                    (short)0, a10, false, false);
        a11 = __builtin_amdgcn_wmma_f32_16x16x4_f32(false, aB, false, b1,
                                                    (short)0, a11, false, false);
    }
    cp0 += 64 * STR;
    cp1 += 64 * STR;
}

// -------------------------------------------------------------------------
// Kernel 1: fused  [x | pe] = adj @ [node_feats-cols | pos]
//   virtual RHS is 8192 x 544 : cols 0..511 -> x[b=j/64][k][i=j%64]
//                               cols 512..543 -> pe[k][j-512]
// Grid: 256 blocks (32 adj rows each).  Block: 544 threads = 17 waves.
// Wave w<16: two node-feat col tiles (stride 64); wave 16: the two pe tiles.
// adj chunk (32x64) double-buffered in LDS; one barrier per chunk.
// -------------------------------------------------------------------------
__global__ __launch_bounds__(544) void k1_aggregate(
    const float* __restrict__ adj,   // [N,N]
    const float* __restrict__ nf,    // [B,N,CIN]
    const float* __restrict__ pos,   // [N,CHID]
    float* __restrict__ xw,          // [B,N,CIN]   (workspace)
    float* __restrict__ pew)         // [N,CHID]    (workspace)
{
    __shared__ float adjS[2 * 32 * 66];      // double buffer, rows padded to 66

    const int tid  = threadIdx.x;
    const int lane = tid & 31;
    const int w    = tid >> 5;               // wave id 0..16
    const int row0 = blockIdx.x * 32;

    const int ct0 = w * 2;                   // column tiles 0..33
    const int ml  = lane & 15;               // A: M index / B: N index
    const int kl  = (lane >> 4) * 2;         // K sub-offset (0 or 2)

    // per-wave B base pointers, pre-offset with the lane term (kl*STR + ml)
    const float* cp0;
    const float* cp1;
    if (w < 16) {
        const int b0 = ct0 >> 2,       i00 = (ct0 & 3) * 16;
        const int b1 = (ct0 + 1) >> 2, i01 = ((ct0 + 1) & 3) * 16;
        cp0 = nf + (size_t)b0 * NN * CIN + i00 + kl * CIN + ml;
        cp1 = nf + (size_t)b1 * NN * CIN + i01 + kl * CIN + ml;
    } else {
        cp0 = pos + 0  + kl * CHID + ml;
        cp1 = pos + 16 + kl * CHID + ml;
    }

    v8f a00 = {}, a01 = {}, a10 = {}, a11 = {};

    // staging lanes: threads 0..511 move one float4 of the adj chunk each
    const int r  = tid >> 4;                 // 0..33 (valid rows: 0..31)
    const int c4 = tid & 15;
    const float* aptr = adj + (size_t)(row0 + r) * NN + c4 * 4;

    float4 pre = {};
    if (tid < 512) pre = *(const float4*)(aptr);

    const int NCHUNK = NN / 64;
    for (int c = 0; c < NCHUNK; ++c) {
        float* buf = &adjS[(c & 1) * (32 * 66)];
        if (tid < 512) {
            float* d = &buf[r * 66 + c4 * 4];
            d[0] = pre.x; d[1] = pre.y; d[2] = pre.z; d[3] = pre.w;
        }
        __syncthreads();                                  // buf ready for all
        if (tid < 512 && (c + 1) < NCHUNK)                // prefetch next chunk
            pre = *(const float4*)(aptr + (size_t)(c + 1) * 64);

        if (w < 16) gemm_chunk<CIN >(buf, cp0, cp1, ml, kl, a00, a01, a10, a11);
        else        gemm_chunk<CHID>(buf, cp0, cp1, ml, kl, a00, a01, a10, a11);
    }

    // ---- store: VGPR v -> M = v (lanes 0-15) / v+8 (lanes 16-31) ----
    float* dst0;
    float* dst1;
    size_t strideO;
    if (w < 16) {
        dst0 = xw + (size_t)(ct0 >> 2) * NN * CIN + (ct0 & 3) * 16 + ml;
        dst1 = xw + (size_t)((ct0 + 1) >> 2) * NN * CIN + ((ct0 + 1) & 3) * 16 + ml;
        strideO = CIN;
    } else {
        dst0 = pew + 0  + ml;
        dst1 = pew + 16 + ml;
        strideO = CHID;
    }
#pragma unroll
    for (int v = 0; v < 8; ++v) {
        const int m = (lane < 16) ? v : v + 8;
        const size_t rA = (size_t)(row0 + m);
        const size_t rB = (size_t)(row0 + 16 + m);
        dst0[rA * strideO] = a00[v];
        dst1[rA * strideO] = a01[v];
        dst0[rB * strideO] = a10[v];
        dst1[rB * strideO] = a11[v];
    }
}

// -------------------------------------------------------------------------
// Kernel 2: per 16-node tile:
//   stage1: pw[16, 64, 64] = pe_tile[16,32] @ W[32, 4096]  -> LDS (256 KB)
//   stage2: out[b, n, :] = x[b, n, :] @ pw[n]  (M=batch padded to 16)
// Block: 256 threads = 8 waves, 288 KB dynamic LDS (CDNA5 WGP: 320 KB).
// -------------------------------------------------------------------------
__global__ __launch_bounds__(256) void k2_final(
    const float* __restrict__ xw,    // [B,N,CIN]
    const float* __restrict__ pew,   // [N,CHID]
    const float* __restrict__ wp,    // [CHID,CIN,COUT]
    float* __restrict__ out)         // [B,N,COUT]
{
    extern __shared__ float smem[];
    float* pwS = smem;                   // [16][64][64] = 65536 floats
    float* xS  = smem + 16 * 64 * 64;    // [8][16][64]  =  8192 floats

    const int tid  = threadIdx.x;
    const int lane = tid & 31;
    const int w    = tid >> 5;           // 0..7
    const int n0   = blockIdx.x * 16;
    const int ml   = lane & 15;
    const int kl   = (lane >> 4) * 2;

    // cooperative load of x tile [8 batches][16 nodes][64] into LDS
#pragma unroll
    for (int j = 0; j < 8; ++j) {
        const int q   = tid + j * 256;   // float4 index 0..2047
        const int b   = q >> 8;
        const int rem = q & 255;
        const int nd  = rem >> 4;
        const int f4  = rem & 15;
        const float4 vv = *(const float4*)(xw + (size_t)b * NN * CIN +
                                           (size_t)(n0 + nd) * CIN + f4 * 4);
        float* d = &xS[(b * 16 + nd) * CIN + f4 * 4];
        d[0] = vv.x; d[1] = vv.y; d[2] = vv.z; d[3] = vv.w;
    }

    // ---- stage 1: pw tile into LDS ----
    const float* peBase = pew + (size_t)(n0 + ml) * CHID;   // A row per lane
    for (int ct = w * 32; ct < w * 32 + 32; ++ct) {         // 16-col tiles of 4096
        const int i  = ct >> 2;
        const int o0 = (ct & 3) * 16;
        const float* wb = wp + (size_t)kl * 4096 + i * 64 + o0 + ml;
        v8f acc = {};
#pragma unroll
        for (int s = 0; s < 8; ++s) {                        // K = 32
            const int k = s * 4;
            v2f a = *(const v2f*)(peBase + k + kl);
            v2f b;
            b.x = wb[(size_t)k * 4096];
            b.y = wb[(size_t)(k + 1) * 4096];
            acc = __builtin_amdgcn_wmma_f32_16x16x4_f32(false, a, false, b,
                                                        (short)0, acc, false, false);
        }
#pragma unroll
        for (int v = 0; v < 8; ++v) {
            const int m = (lane < 16) ? v : v + 8;           // node in tile
            pwS[(m * 64 + i) * 64 + o0 + ml] = acc[v];
        }
    }
    __syncthreads();

    // ---- stage 2: out[:, nd, :] = x_nd(8x64) @ pw_nd(64x64) ----
    for (int nn = 0; nn < 2; ++nn) {
        const int nd = w * 2 + nn;
#pragma unroll
        for (int oc = 0; oc < 4; ++oc) {
            v8f acc = {};
#pragma unroll
            for (int s = 0; s < 16; ++s) {                   // K = 64
                const int k = s * 4 + kl;
                v2f a;
                if (ml < 8) {                                // M rows 8..15 zero pad
                    a = *(const v2f*)(&xS[(ml * 16 + nd) * CIN + k]);
                } else {
                    a.x = 0.0f; a.y = 0.0f;
                }
                v2f b;
                b.x = pwS[(nd * 64 + k) * 64 + oc * 16 + ml];
                b.y = pwS[(nd * 64 + k + 1) * 64 + oc * 16 + ml];
                acc = __builtin_amdgcn_wmma_f32_16x16x4_f32(false, a, false, b,
                                                            (short)0, acc, false, false);
            }
            if (lane < 16) {                                 // M = v = batch (0..7)
#pragma unroll
                for (int v = 0; v < 8; ++v) {
                    out[(size_t)v * NN * COUT + (size_t)(n0 + nd) * COUT +
                        oc * 16 + ml] = acc[v];
                }
            }
        }
    }
}

// -------------------------------------------------------------------------
extern "C" void kernel_launch(void* const* d_in, const int* in_sizes, int n_in,
                              void* d_out, int out_size, void* d_ws, size_t ws_size,
                              hipStream_t stream) {
    const float* nf  = (const float*)d_in[0];   // node_feats      [8,8192,64]
    const float* adj = (const float*)d_in[1];   // adj_matrix      [8192,8192]
    const float* pos = (const float*)d_in[2];   // position_embed  [8192,32]
    const float* wp  = (const float*)d_in[3];   // weights_pool    [32,64,64]
    float* outp = (float*)d_out;

    float* xw  = (float*)d_ws;                              // 8*8192*64 floats (16 MB)
    float* pew = xw + (size_t)BATCH * NN * CIN;             // 8192*32 floats  (1 MB)

    k1_aggregate<<<dim3(NN / 32), dim3(544), 0, stream>>>(adj, nf, pos, xw, pew);

    const size_t smem2 = (size_t)(16 * 64 * 64 + 8 * 16 * 64) * sizeof(float); // 288 KB
    (void)hipFuncSetAttribute((const void*)k2_final,
                              hipFuncAttributeMaxDynamicSharedMemorySize, (int)smem2);
    k2_final<<<dim3(NN / 16), dim3(256), smem2, stream>>>(xw, pew, wp, outp);
}